// GCN_71975061946861
// MI455X (gfx1250) — compile-verified
//
#include <hip/hip_runtime.h>
#include <hip/hip_bf16.h>

typedef __attribute__((ext_vector_type(2))) float v2f;
typedef __attribute__((ext_vector_type(8))) float v8f;

// ---------------------------------------------------------------------------
// Degree / normalization precompute
// ---------------------------------------------------------------------------
__global__ __launch_bounds__(256) void gcn_init_deg(float* __restrict__ deg, int n) {
    int i = blockIdx.x * blockDim.x + threadIdx.x;
    if (i < n) deg[i] = 1.0f;  // self-loop
}

__global__ __launch_bounds__(256) void gcn_count_deg(const int* __restrict__ dst,
                                                     float* __restrict__ deg, int e) {
    int i = blockIdx.x * blockDim.x + threadIdx.x;
    if (i < e) unsafeAtomicAdd(&deg[dst[i]], 1.0f);
}

__global__ __launch_bounds__(256) void gcn_rsqrt_deg(float* __restrict__ deg, int n) {
    int i = blockIdx.x * blockDim.x + threadIdx.x;
    if (i < n) deg[i] = rsqrtf(deg[i]);  // in-place: deg -> dinv
}

__global__ __launch_bounds__(256) void gcn_edge_norm(const int* __restrict__ src,
                                                     const int* __restrict__ dst,
                                                     const float* __restrict__ dinv,
                                                     float* __restrict__ normw, int e) {
    int i = blockIdx.x * blockDim.x + threadIdx.x;
    if (i < e) normw[i] = dinv[src[i]] * dinv[dst[i]];
}

// ---------------------------------------------------------------------------
// Dense transform h = x @ W via V_WMMA_F32_16X16X4_F32 (f32 exact path).
// Also seeds agg with the self-loop contribution h[i] * dinv[i]^2.
// One wave per 16x16 output tile; B (16-col slice of W) hoisted into regs.
// Requires n % 16 == 0 (50000 = 3125 * 16).
// ---------------------------------------------------------------------------
__global__ __launch_bounds__(256)
void gcn_gemm_wmma(const float* __restrict__ x, const float* __restrict__ W,
                   const float* __restrict__ dinv,
                   float* __restrict__ h, float* __restrict__ agg, int n) {
    const int lane = threadIdx.x & 31;
    const int wave = (blockIdx.x * blockDim.x + threadIdx.x) >> 5;
    const int totalWaves = (gridDim.x * blockDim.x) >> 5;
    const int ct   = wave & 3;    // column tile 0..3 (DIM=64 -> 4 tiles)
    const int half = lane >> 4;   // lane half: selects K-pair / row-half
    const int l16  = lane & 15;
    const int rowTiles = n >> 4;

    // Preload B fragments for this column tile: B[k][ct*16 + l16]
    // Layout: VGPR0 = row (k0 + 2*half), VGPR1 = row (k0 + 2*half + 1)
    v2f bfrag[16];
#pragma unroll
    for (int ks = 0; ks < 16; ++ks) {
        const int k = ks * 4 + 2 * half;
        bfrag[ks].x = W[(size_t)(k)     * 64 + ct * 16 + l16];
        bfrag[ks].y = W[(size_t)(k + 1) * 64 + ct * 16 + l16];
    }

    for (int rt = wave >> 2; rt < rowTiles; rt += (totalWaves >> 2)) {
        const int m0   = rt << 4;
        const int arow = m0 + l16;  // A: lane holds row M = lane&15
        v8f c = {};
#pragma unroll
        for (int ks = 0; ks < 16; ++ks) {
            const int k = ks * 4 + 2 * half;
            // A 16x4 f32 fragment: {x[arow][k], x[arow][k+1]} (8B-aligned)
            v2f a = *(const v2f*)(x + (size_t)arow * 64 + k);
            c = __builtin_amdgcn_wmma_f32_16x16x4_f32(
                    /*neg_a=*/false, a, /*neg_b=*/false, bfrag[ks],
                    /*c_mod=*/(short)0, c, /*reuse_a=*/false, /*reuse_b=*/false);
        }
        // D layout: VGPR v -> row m0 + v + 8*half, col = ct*16 + (lane&15)
        const int col = ct * 16 + l16;
#pragma unroll
        for (int v = 0; v < 8; ++v) {
            const int row = m0 + v + 8 * half;
            const float val = c[v];
            h[(size_t)row * 64 + col] = val;
            const float di = dinv[row];
            agg[(size_t)row * 64 + col] = val * di * di;  // self-loop term
        }
    }
}

// ---------------------------------------------------------------------------
// Edge scatter: agg[dst] += h[src] * norm.  One wave per edge, 2 feats/lane.
// ---------------------------------------------------------------------------
__global__ __launch_bounds__(256)
void gcn_scatter(const float* __restrict__ h, const int* __restrict__ src,
                 const int* __restrict__ dst, const float* __restrict__ normw,
                 float* __restrict__ agg, int nEdges) {
    const long long gid = (long long)blockIdx.x * blockDim.x + threadIdx.x;
    const int e = (int)(gid >> 5);
    const int r = (int)(gid & 31);
    if (e >= nEdges) return;
    const int s   = src[e];
    const int d   = dst[e];
    const float w = normw[e];
    const v2f v = *(const v2f*)(h + (size_t)s * 64 + r * 2);
    float* p = agg + (size_t)d * 64 + r * 2;
    unsafeAtomicAdd(p,     v.x * w);
    unsafeAtomicAdd(p + 1, v.y * w);
}

// ---------------------------------------------------------------------------
// bias + LayerNorm + ReLU.  One wave32 per node, 2 features/lane,
// shuffle-based reduction (wave32 native on gfx1250).
// ---------------------------------------------------------------------------
__global__ __launch_bounds__(256)
void gcn_post_ln_relu(const float* __restrict__ agg, const float* __restrict__ bias,
                      const float* __restrict__ gamma, const float* __restrict__ beta,
                      float* __restrict__ out, int n) {
    const int node = (blockIdx.x * blockDim.x + threadIdx.x) >> 5;
    const int lane = threadIdx.x & 31;
    if (node >= n) return;
    const int f = lane * 2;
    v2f v = *(const v2f*)(agg + (size_t)node * 64 + f);
    v.x += bias[f];
    v.y += bias[f + 1];
    float s  = v.x + v.y;
    float sq = v.x * v.x + v.y * v.y;
#pragma unroll
    for (int off = 16; off >= 1; off >>= 1) {
        s  += __shfl_xor(s, off, 32);
        sq += __shfl_xor(sq, off, 32);
    }
    const float mean = s * (1.0f / 64.0f);
    const float var  = sq * (1.0f / 64.0f) - mean * mean;
    const float rinv = rsqrtf(var + 1e-5f);
    float ox = (v.x - mean) * rinv * gamma[f]     + beta[f];
    float oy = (v.y - mean) * rinv * gamma[f + 1] + beta[f + 1];
    v2f o;
    o.x = fmaxf(ox, 0.0f);
    o.y = fmaxf(oy, 0.0f);
    *(v2f*)(out + (size_t)node * 64 + f) = o;
}

// ---------------------------------------------------------------------------
// Host launcher
// ---------------------------------------------------------------------------
extern "C" void kernel_launch(void* const* d_in, const int* in_sizes, int n_in,
                              void* d_out, int out_size, void* d_ws, size_t ws_size,
                              hipStream_t stream) {
    const float* x      = (const float*)d_in[0];   // [N,64]
    const int*   ei     = (const int*)d_in[1];     // [2,E] flat
    const float* Ws     = (const float*)d_in[2];   // [3,64,64]
    const float* bs     = (const float*)d_in[3];   // [3,64]
    const float* gammas = (const float*)d_in[4];   // [3,64]
    const float* betas  = (const float*)d_in[5];   // [3,64]

    const int n = in_sizes[0] / 64;   // 50000 (multiple of 16)
    const int E = in_sizes[1] / 2;    // 800000
    const int* src = ei;
    const int* dst = ei + E;

    auto align256 = [](size_t v) { return (v + 255) & ~(size_t)255; };
    char* ws = (char*)d_ws;
    float* h     = (float*)ws; ws += align256((size_t)n * 64 * sizeof(float));
    float* agg   = (float*)ws; ws += align256((size_t)n * 64 * sizeof(float));
    float* xbuf  = (float*)ws; ws += align256((size_t)n * 64 * sizeof(float));
    float* dinv  = (float*)ws; ws += align256((size_t)n * sizeof(float));
    float* normw = (float*)ws;

    dim3 blk(256);
    const int gN = (n + 255) / 256;
    const int gE = (E + 255) / 256;

    // Precompute (once, reused across all 3 layers)
    gcn_init_deg<<<gN, blk, 0, stream>>>(dinv, n);
    gcn_count_deg<<<gE, blk, 0, stream>>>(dst, dinv, E);
    gcn_rsqrt_deg<<<gN, blk, 0, stream>>>(dinv, n);
    gcn_edge_norm<<<gE, blk, 0, stream>>>(src, dst, dinv, normw, E);

    // 3125 row tiles * 4 col tiles = 12500 waves; 8 waves/block
    const int gemmGrid = ((n >> 4) * 4 + 7) / 8;            // 1563
    const int scatterGrid = (int)(((long long)E * 32 + 255) / 256);
    const int postGrid = (int)(((long long)n * 32 + 255) / 256);

    const float* xin = x;
    for (int i = 0; i < 3; ++i) {
        const float* W = Ws + (size_t)i * 64 * 64;
        gcn_gemm_wmma<<<gemmGrid, blk, 0, stream>>>(xin, W, dinv, h, agg, n);
        gcn_scatter<<<scatterGrid, blk, 0, stream>>>(h, src, dst, normw, agg, E);
        float* xout = (i == 2) ? (float*)d_out : xbuf;
        gcn_post_ln_relu<<<postGrid, blk, 0, stream>>>(
            agg, bs + i * 64, gammas + i * 64, betas + i * 64, xout, n);
        xin = xout;
    }
}